// ElectricOverflow_27650999452253
// MI455X (gfx1250) — compile-verified
//
#include <hip/hip_runtime.h>

// ---------------- problem constants (match reference) ----------------
constexpr int   NM   = 400000;
constexpr int   NT   = 50000;
constexpr int   NF   = 500000;
constexpr int   NTOT = NM + NT + NF;      // 950000
constexpr int   NBX  = 1024;
constexpr int   NBY  = 1024;
constexpr int   NBINS = NBX * NBY;
constexpr float XL = 0.0f, YL = 0.0f;
constexpr float BSX = 1.0f, BSY = 1.0f;
constexpr float TD = 0.9f;
constexpr float BIN_AREA = BSX * BSY;
constexpr float TDBA = TD * BIN_AREA;     // 0.9
constexpr int   K_MOV = 5;
constexpr int   K_FIX = 6;

// ---------------- WMMA wave32 sum ----------------
typedef float v2f __attribute__((ext_vector_type(2)));
typedef float v8f __attribute__((ext_vector_type(8)));

// Full-wave (32 lane) sum using V_WMMA_F32_16X16X4_F32.
// A(16x4): lane m<16 -> {v(m), 0} in K=0,1 ; lane m+16 -> {v(m+16), 0} in K=2,3.
// B = all-ones. D[i][j] = rowsum(i) = v(i) + v(i+16).
// After WMMA, lanes 0-15 hold rows 0-7 in c[0..7], lanes 16-31 hold rows 8-15.
// Summing c[0..7] gives half-sums; a second WMMA pairs the two halves -> total.
// REQUIRES EXEC == all ones (caller guarantees no divergence).
__device__ __forceinline__ float wave_sum_wmma(float v) {
  v2f ones; ones[0] = 1.0f; ones[1] = 1.0f;
  v2f a;    a[0] = v;       a[1] = 0.0f;
  v8f c = {};
  c = __builtin_amdgcn_wmma_f32_16x16x4_f32(false, a, false, ones,
                                            (short)0, c, false, false);
  float s = (c[0] + c[1]) + (c[2] + c[3]) + (c[4] + c[5]) + (c[6] + c[7]);
  v2f a2;  a2[0] = s;  a2[1] = 0.0f;
  v8f c2 = {};
  c2 = __builtin_amdgcn_wmma_f32_16x16x4_f32(false, a2, false, ones,
                                             (short)0, c2, false, false);
  return c2[0];  // total on every lane
}

// ---------------- kernel 1: zero map + outputs ----------------
__global__ void eo_zero_kernel(float* __restrict__ map, float* __restrict__ out) {
  int idx = blockIdx.x * blockDim.x + threadIdx.x;
  if (idx < NBINS) map[idx] = 0.0f;
  if (idx == 0) { out[0] = 0.0f; out[1] = 0.0f; }
}

// ---------------- kernel 2: terminal (fixed) 6x6 splat, weight = TD ----------------
__global__ void eo_fixed_scatter(const float* __restrict__ pos,
                                 const float* __restrict__ nsx,
                                 const float* __restrict__ nsy,
                                 float* __restrict__ map) {
  int i = blockIdx.x * blockDim.x + threadIdx.x;
  if (i >= NT) return;
  int n = NM + i;
  float x = pos[n];
  float y = pos[NTOT + n];
  float lox = fminf(fmaxf(x, XL), XL + (float)NBX * BSX);
  float hix = fminf(fmaxf(x + nsx[n], XL), XL + (float)NBX * BSX);
  float loy = fminf(fmaxf(y, YL), YL + (float)NBY * BSY);
  float hiy = fminf(fmaxf(y + nsy[n], YL), YL + (float)NBY * BSY);
  float szx = hix - lox;
  float szy = hiy - loy;
  int bx0 = (int)floorf((lox - XL) / BSX);
  int by0 = (int)floorf((loy - YL) / BSY);

  float py[K_FIX];
#pragma unroll
  for (int k = 0; k < K_FIX; ++k) {
    int ky = by0 + k;
    float bl = YL + (float)ky * BSY;
    float p = fminf(loy + szy, bl + BSY) - fmaxf(loy, bl);
    py[k] = (ky >= 0 && ky < NBY) ? fmaxf(p, 0.0f) : 0.0f;
  }
#pragma unroll
  for (int kx = 0; kx < K_FIX; ++kx) {
    int bx = bx0 + kx;
    float bl = XL + (float)bx * BSX;
    float px = fminf(lox + szx, bl + BSX) - fmaxf(lox, bl);
    px = (bx >= 0 && bx < NBX) ? fmaxf(px, 0.0f) : 0.0f;
    if (px <= 0.0f) continue;
    float wpx = TD * px;
    int rowbase = bx * NBY + by0;
#pragma unroll
    for (int ky = 0; ky < K_FIX; ++ky) {
      float v = wpx * py[ky];
      if (v > 0.0f) atomicAdd(map + rowbase + ky, v);
    }
  }
}

// ---------------- kernel 3: movable + filler 5x5 splat, weight = ratio ----------------
__global__ void eo_mov_scatter(const float* __restrict__ pos,
                               const float* __restrict__ nsxc,
                               const float* __restrict__ nsyc,
                               const float* __restrict__ offx,
                               const float* __restrict__ offy,
                               const float* __restrict__ ratio,
                               float* __restrict__ map) {
  int i = blockIdx.x * blockDim.x + threadIdx.x;
  if (i >= NM + NF) return;
  int n = (i < NM) ? i : (NM + NT + (i - NM));  // skip terminal range
  float lx = pos[n] + offx[n];
  float ly = pos[NTOT + n] + offy[n];
  float sx = nsxc[n];
  float sy = nsyc[n];
  float w  = ratio[n];
  int bx0 = (int)floorf((lx - XL) / BSX);
  int by0 = (int)floorf((ly - YL) / BSY);

  float py[K_MOV];
#pragma unroll
  for (int k = 0; k < K_MOV; ++k) {
    int ky = by0 + k;
    float bl = YL + (float)ky * BSY;
    float p = fminf(ly + sy, bl + BSY) - fmaxf(ly, bl);
    py[k] = (ky >= 0 && ky < NBY) ? fmaxf(p, 0.0f) : 0.0f;
  }
#pragma unroll
  for (int kx = 0; kx < K_MOV; ++kx) {
    int bx = bx0 + kx;
    float bl = XL + (float)bx * BSX;
    float px = fminf(lx + sx, bl + BSX) - fmaxf(lx, bl);
    px = (bx >= 0 && bx < NBX) ? fmaxf(px, 0.0f) : 0.0f;
    if (px <= 0.0f) continue;
    float wpx = w * px;
    int rowbase = bx * NBY + by0;
#pragma unroll
    for (int ky = 0; ky < K_MOV; ++ky) {
      float v = wpx * py[ky];
      if (v > 0.0f) atomicAdd(map + rowbase + ky, v);
    }
  }
}

// ---------------- kernel 4: reduce map -> (cost, max) ----------------
// Launched with grid*block == NBINS exactly: no divergent tail, so EXEC is
// all-1s at the WMMA (required by CDNA5 WMMA restrictions).
__global__ void eo_reduce_kernel(const float* __restrict__ map,
                                 const unsigned char* __restrict__ mask,
                                 float* __restrict__ out) {
  int idx = blockIdx.x * blockDim.x + threadIdx.x;
  float v = map[idx];
  if (mask[idx]) v = TDBA;              // padding_mask override (cndmask, no branch)
  float cost = fmaxf(v - TDBA, 0.0f);
  float vmax = v;

  // wave32 sum via matrix unit (v_wmma_f32_16x16x4_f32 x2)
  float wsum = wave_sum_wmma(cost);

  // wave32 max via lane shuffles
#pragma unroll
  for (int off = 16; off >= 1; off >>= 1)
    vmax = fmaxf(vmax, __shfl_xor(vmax, off, 32));

  if ((threadIdx.x & 31) == 0) {
    atomicAdd(&out[0], wsum);
    // densities are >= 0, so integer max over IEEE bit patterns == float max
    atomicMax((int*)&out[1], __float_as_int(vmax));
  }
}

// ---------------- launcher ----------------
extern "C" void kernel_launch(void* const* d_in, const int* in_sizes, int n_in,
                              void* d_out, int out_size, void* d_ws, size_t ws_size,
                              hipStream_t stream) {
  const float* pos   = (const float*)d_in[0];          // 2N
  const float* nsx   = (const float*)d_in[1];          // N
  const float* nsy   = (const float*)d_in[2];          // N
  const float* nsxc  = (const float*)d_in[3];          // N
  const float* nsyc  = (const float*)d_in[4];          // N
  const float* offx  = (const float*)d_in[5];          // N
  const float* offy  = (const float*)d_in[6];          // N
  const float* ratio = (const float*)d_in[7];          // N
  const unsigned char* mask = (const unsigned char*)d_in[8]; // NBX*NBY bool

  float* out = (float*)d_out;                          // [cost, max/BIN_AREA]
  float* map = (float*)d_ws;                           // NBINS floats (4 MB)

  (void)in_sizes; (void)n_in; (void)out_size; (void)ws_size;

  eo_zero_kernel<<<(NBINS + 255) / 256, 256, 0, stream>>>(map, out);
  eo_fixed_scatter<<<(NT + 255) / 256, 256, 0, stream>>>(pos, nsx, nsy, map);
  eo_mov_scatter<<<(NM + NF + 255) / 256, 256, 0, stream>>>(
      pos, nsxc, nsyc, offx, offy, ratio, map);
  eo_reduce_kernel<<<NBINS / 256, 256, 0, stream>>>(map, mask, out);
}